// Inter_fuse_module_36395552866670
// MI455X (gfx1250) — compile-verified
//
#include <hip/hip_runtime.h>
#include <math.h>

typedef float v2f __attribute__((ext_vector_type(2)));
typedef float v8f __attribute__((ext_vector_type(8)));

#define CDIM 64
#define LDS_STRIDE 68   // pad 64 -> 68 to avoid 16-way LDS bank conflicts on column reads

// ---- order-preserving float <-> uint encoding for atomic segment-max ----
__device__ __forceinline__ unsigned enc_f(float f) {
  unsigned u = __float_as_uint(f);
  return (u & 0x80000000u) ? ~u : (u | 0x80000000u);
}
__device__ __forceinline__ float dec_f(unsigned u) {
  return (u & 0x80000000u) ? __uint_as_float(u ^ 0x80000000u) : __uint_as_float(~u);
}
#define ENC_NEG_INF 0x007FFFFFu   // enc(-inf)

// ---------------- kernel 0: per-call re-init of stateful scratch ----------------
__global__ void init_kernel(unsigned* __restrict__ amaxu, float* __restrict__ denom,
                            double* __restrict__ scal, int n) {
  int i = blockIdx.x * blockDim.x + threadIdx.x;
  if (i < n) { amaxu[i] = ENC_NEG_INF; denom[i] = 0.0f; }
  if (i == 0) { scal[0] = 0.0; scal[1] = 0.0; }
}

// One 16x16 output tile: D = A(16x64, LDS) * W^T(16 cols) + bias, via 16 chained
// fp32 WMMAs. Direct pointer args keep address-space inference intact so the
// compiler emits global_load/global_store (not flat_*) for W/bias/out.
__device__ __forceinline__ void gemm_tile(
    const float* __restrict__ A,      // &lds[tile][(rt*16+m)*LDS_STRIDE]
    const float* __restrict__ W,      // [64,64] row-major (out x in)
    const float* __restrict__ bias,   // [64]
    float* __restrict__ out,          // [N,64]
    int rowBase, int colBase, int m, int khalf, int rowHi) {
  const float* Brow = W + (size_t)(colBase + m) * CDIM;   // B[k][n] = W[n][k]
  v8f c = {};
#pragma unroll
  for (int kk = 0; kk < CDIM; kk += 4) {
    v2f a, b;
    a[0] = A[kk + khalf];     a[1] = A[kk + khalf + 1];
    b[0] = Brow[kk + khalf];  b[1] = Brow[kk + khalf + 1];
    c = __builtin_amdgcn_wmma_f32_16x16x4_f32(false, a, false, b, (short)0, c, false, false);
  }
  const float bv = bias[colBase + m];
#pragma unroll
  for (int r = 0; r < 8; ++r) {
    out[(size_t)(rowBase + rowHi + r) * CDIM + colBase + m] = c[r] + bv;
  }
}

// ---------------- kernel 1: fused q/k/v/skip GEMMs via fp32 WMMA ----------------
// Block: 256 threads = 8 waves; handles 32 rows x 64 cols for all 4 matrices.
__global__ __launch_bounds__(256) void qkvs_gemm_kernel(
    const float* __restrict__ geo, const float* __restrict__ euc,
    const float* __restrict__ Wq, const float* __restrict__ bq,
    const float* __restrict__ Wk, const float* __restrict__ bk,
    const float* __restrict__ Wv, const float* __restrict__ bv,
    const float* __restrict__ Ws, const float* __restrict__ bs,
    float* __restrict__ qb, float* __restrict__ kb,
    float* __restrict__ vb, float* __restrict__ ob, int n) {
  __shared__ float ldsE[32 * LDS_STRIDE];   // euc tile
  __shared__ float ldsG[32 * LDS_STRIDE];   // geo tile

  const int t = threadIdx.x;
  const int rowBase0 = blockIdx.x * 32;

  // stage 32x64 tiles of euc and geo into LDS (coalesced float4 loads)
  for (int c = t; c < 512; c += 256) {            // 512 float4 chunks per tile
    const int row  = c >> 4;
    const int col4 = (c & 15) << 2;
    int grow = rowBase0 + row;
    if (grow >= n) grow = n - 1;                  // N is a multiple of 32 anyway
    const size_t goff = (size_t)grow * CDIM + col4;
    const float4 ve = *(const float4*)(euc + goff);
    const float4 vg = *(const float4*)(geo + goff);
    float* de = &ldsE[row * LDS_STRIDE + col4];
    float* dg = &ldsG[row * LDS_STRIDE + col4];
    de[0] = ve.x; de[1] = ve.y; de[2] = ve.z; de[3] = ve.w;
    dg[0] = vg.x; dg[1] = vg.y; dg[2] = vg.z; dg[3] = vg.w;
  }
  __syncthreads();

  const int lane  = t & 31;
  const int wave  = t >> 5;
  const int rt    = wave >> 2;          // 0..1
  const int ct    = wave & 3;           // 0..3
  const int rowBase = rowBase0 + rt * 16;
  const int colBase = ct * 16;
  const int m     = lane & 15;          // M (A) / N (B,C,D) index
  const int khalf = (lane >> 4) << 1;   // K offset: lanes 16-31 carry K+2 (ISA 16x4 layout)
  const int rowHi = (lane >> 4) ? 8 : 0;

  const float* AE = &ldsE[(rt * 16 + m) * LDS_STRIDE];
  const float* AG = &ldsG[(rt * 16 + m) * LDS_STRIDE];

  gemm_tile(AE, Wq, bq, qb, rowBase, colBase, m, khalf, rowHi);  // q from euc
  gemm_tile(AG, Wk, bk, kb, rowBase, colBase, m, khalf, rowHi);  // k from geo
  gemm_tile(AG, Wv, bv, vb, rowBase, colBase, m, khalf, rowHi);  // v from geo
  gemm_tile(AE, Ws, bs, ob, rowBase, colBase, m, khalf, rowHi);  // skip from euc
}

// ---------------- kernel 2: per-edge logits + segment max ----------------
__global__ __launch_bounds__(256) void edge_alpha_kernel(
    const long long* __restrict__ ei, const float* __restrict__ qb,
    const float* __restrict__ kb, float* __restrict__ alpha,
    unsigned* __restrict__ amaxu, long long E) {
  const int lane = threadIdx.x & 31;
  const long long e = (long long)blockIdx.x * 8 + (threadIdx.x >> 5);
  if (e >= E) return;
  const int s = (int)ei[e];
  const int d = (int)ei[E + e];
  const float2 qv = *(const float2*)(qb + (size_t)d * CDIM + lane * 2);
  const float2 kv = *(const float2*)(kb + (size_t)s * CDIM + lane * 2);
  float p = qv.x * kv.x + qv.y * kv.y;
#pragma unroll
  for (int off = 16; off > 0; off >>= 1) p += __shfl_xor(p, off, 32);
  if (lane == 0) {
    const float a = p * 0.125f;   // / sqrt(64)
    alpha[e] = a;
    atomicMax(&amaxu[d], enc_f(a));
  }
}

// ---------------- kernel 3: exp(alpha - amax) + segment sum ----------------
__global__ __launch_bounds__(256) void edge_exp_kernel(
    const long long* __restrict__ ei, float* __restrict__ alpha,
    const unsigned* __restrict__ amaxu, float* __restrict__ denom, long long E) {
  const long long e = (long long)blockIdx.x * blockDim.x + threadIdx.x;
  if (e >= E) return;
  const int d = (int)ei[E + e];
  float mx = dec_f(amaxu[d]);
  if (!__builtin_isfinite(mx)) mx = 0.0f;   // reference: non-finite amax -> 0
  const float ex = __expf(alpha[e] - mx);
  alpha[e] = ex;
  atomicAdd(&denom[d], ex);
}

// ---------------- kernel 4: weighted scatter-aggregate of v ----------------
__global__ __launch_bounds__(256) void edge_agg_kernel(
    const long long* __restrict__ ei, const float* __restrict__ alpha,
    const float* __restrict__ denom, const float* __restrict__ vb,
    float* __restrict__ ob, long long E) {
  const int lane = threadIdx.x & 31;
  const long long e = (long long)blockIdx.x * 8 + (threadIdx.x >> 5);
  if (e >= E) return;
  const int s = (int)ei[e];
  const int d = (int)ei[E + e];
  const float w = alpha[e] / (denom[d] + 1e-16f);
  const float2 vv = *(const float2*)(vb + (size_t)s * CDIM + lane * 2);
  float* o = ob + (size_t)d * CDIM + lane * 2;
  atomicAdd(o,     w * vv.x);
  atomicAdd(o + 1, w * vv.y);
}

// ---------------- kernel 5: global sum / sum-of-squares (graph LayerNorm) ----------------
__global__ __launch_bounds__(256) void reduce_kernel(
    const float* __restrict__ ob, double* __restrict__ scal, long long total) {
  double s = 0.0, s2 = 0.0;
  for (long long i = (long long)blockIdx.x * blockDim.x + threadIdx.x; i < total;
       i += (long long)gridDim.x * blockDim.x) {
    const float x = ob[i];
    s  += x;
    s2 += (double)x * (double)x;
  }
#pragma unroll
  for (int off = 16; off > 0; off >>= 1) {
    s  += __shfl_xor(s,  off, 32);
    s2 += __shfl_xor(s2, off, 32);
  }
  if ((threadIdx.x & 31) == 0) {
    atomicAdd(&scal[0], s);
    atomicAdd(&scal[1], s2);
  }
}

// ---------------- kernel 6: normalize + affine + ReLU ----------------
__global__ __launch_bounds__(256) void finalize_kernel(
    const float* __restrict__ ob, const double* __restrict__ scal,
    const float* __restrict__ lnw, const float* __restrict__ lnb,
    float* __restrict__ out, long long total, double inv_cnt) {
  const long long i = (long long)blockIdx.x * blockDim.x + threadIdx.x;
  if (i >= total) return;
  const double mean_d = scal[0] * inv_cnt;
  const double ex2_d  = scal[1] * inv_cnt;
  const float mean = (float)mean_d;
  float var = (float)(ex2_d - mean_d * mean_d);   // E[x^2]-mean^2 == mean(xc^2)
  var = fmaxf(var, 0.0f);
  const float stdv = sqrtf(var);
  const int c = (int)(i & (CDIM - 1));
  const float y = (ob[i] - mean) / (stdv + 1e-5f) * lnw[c] + lnb[c];
  out[i] = fmaxf(y, 0.0f);
}

extern "C" void kernel_launch(void* const* d_in, const int* in_sizes, int n_in,
                              void* d_out, int out_size, void* d_ws, size_t ws_size,
                              hipStream_t stream) {
  const float* geo = (const float*)d_in[0];
  const float* euc = (const float*)d_in[1];
  const float* Wq  = (const float*)d_in[2];
  const float* bq  = (const float*)d_in[3];
  const float* Wk  = (const float*)d_in[4];
  const float* bk  = (const float*)d_in[5];
  const float* Wv  = (const float*)d_in[6];
  const float* bv  = (const float*)d_in[7];
  const float* Ws  = (const float*)d_in[8];
  const float* bs  = (const float*)d_in[9];
  const float* lnw = (const float*)d_in[10];
  const float* lnb = (const float*)d_in[11];
  const long long* ei = (const long long*)d_in[12];   // int64 edge_index [2, E]

  const int       N  = in_sizes[0] / CDIM;
  const long long E  = (long long)in_sizes[12] / 2;
  const size_t    NC = (size_t)N * CDIM;

  // workspace layout (f32 unless noted)
  float*    qb    = (float*)d_ws;
  float*    kb    = qb + NC;
  float*    vb    = kb + NC;
  float*    ob    = vb + NC;          // skip output, then agg accumulates into it
  float*    alpha = ob + NC;          // E (logits, then exp weights)
  float*    denom = alpha + E;        // N
  unsigned* amaxu = (unsigned*)(denom + N);   // N
  double*   scal  = (double*)(amaxu + N);     // 2 doubles (8B-aligned offset)

  float* out = (float*)d_out;

  init_kernel<<<(N + 255) / 256, 256, 0, stream>>>(amaxu, denom, scal, N);

  qkvs_gemm_kernel<<<(N + 31) / 32, 256, 0, stream>>>(
      geo, euc, Wq, bq, Wk, bk, Wv, bv, Ws, bs, qb, kb, vb, ob, N);

  edge_alpha_kernel<<<(int)((E + 7) / 8), 256, 0, stream>>>(ei, qb, kb, alpha, amaxu, E);

  edge_exp_kernel<<<(int)((E + 255) / 256), 256, 0, stream>>>(ei, alpha, amaxu, denom, E);

  edge_agg_kernel<<<(int)((E + 7) / 8), 256, 0, stream>>>(ei, alpha, denom, vb, ob, E);

  reduce_kernel<<<1024, 256, 0, stream>>>(ob, scal, (long long)NC);

  finalize_kernel<<<(int)((NC + 255) / 256), 256, 0, stream>>>(
      ob, scal, lnw, lnb, out, (long long)NC, 1.0 / (double)NC);
}